// MambaEncoder_38259568673463
// MI455X (gfx1250) — compile-verified
//
#include <hip/hip_runtime.h>
#include <hip/hip_bf16.h>
#include <math.h>

typedef __attribute__((ext_vector_type(2))) float v2f;
typedef __attribute__((ext_vector_type(8))) float v8f;

#define B_   2
#define L_   2048
#define F_   128
#define DM_  768
#define NL_  4
#define DS_  16
#define DC_  4
#define DI_  1536
#define DR_  48
#define ROWS (B_*L_)        /* 4096 */
#define XZW  (2*DI_)        /* 3072 */
#define DBW  (DR_ + 2*DS_)  /* 80   */

__device__ __forceinline__ float sigmoidf_(float x){ return 1.f/(1.f+expf(-x)); }

// ---------------------------------------------------------------------------
// fp32 WMMA GEMM: D = epilogue(A[M,K] * B[K,N] (+bias) (+Cin))
// One wave computes a 16 x (NT*16) strip. K is processed in chunks of 16:
// all fragments for 4 WMMA k-steps are loaded first (one load clause, one
// wait), then 4*NT back-to-back v_wmma_f32_16x16x4_f32. Next chunk is
// prefetched (global_prefetch_b8) while the matrix pipe drains.
// A frag (16x4 f32): lane<16 -> row=lane, K={k,k+1}; lane>=16 -> K={k+2,k+3}.
// B frag (4x16):     v0 = row k (+2 hi half), v1 = row k+1 (+3 hi half).
// C/D (16x16):       vgpr r -> row r (lanes 0-15) / r+8 (lanes 16-31).
// EPI: 0=store, 1=+bias, 2=softplus(x+bias), 3=accumulate into Cin
// ---------------------------------------------------------------------------
template<int NT, int EPI>
__global__ void gemm_wmma_f32(const float* __restrict__ A, const float* __restrict__ Bm,
                              const float* __restrict__ bias, const float* __restrict__ Cin,
                              float* __restrict__ D,
                              int K, int lda, int ldb, int ldd,
                              int tilesM, int totalWaves)
{
    int gw = (int)((blockIdx.x * blockDim.x + threadIdx.x) >> 5);
    if (gw >= totalWaves) return;              // wave-uniform: EXEC stays all-1
    int lane = threadIdx.x & 31;
    int tm = gw % tilesM;
    int tg = gw / tilesM;
    int m0 = tm << 4;
    int n0 = tg * (NT * 16);
    int mr = lane & 15;                        // row (A) / col (B,D)
    int kh = (lane >> 4) << 1;                 // K sub-offset: 0 or 2

    const float* Ap = A  + (size_t)(m0 + mr) * lda + kh;
    const float* Bp = Bm + (size_t)kh * ldb + n0 + mr;
    const size_t bRow   = (size_t)ldb;
    const size_t bStep4 = (size_t)ldb * 4;
    const size_t bChunk = (size_t)ldb * 16;

    v8f acc[NT];
#pragma unroll
    for (int j = 0; j < NT; ++j) acc[j] = (v8f)0.f;

    for (int k = 0; k < K; k += 16) {
        v2f a[4];
        v2f bfr[4][NT];
        const float* Bk = Bp;
#pragma unroll
        for (int ks = 0; ks < 4; ++ks) {
            a[ks].x = Ap[4 * ks];
            a[ks].y = Ap[4 * ks + 1];
#pragma unroll
            for (int j = 0; j < NT; ++j) {
                bfr[ks][j].x = Bk[j * 16];
                bfr[ks][j].y = Bk[j * 16 + bRow];
            }
            Bk += bStep4;
        }
        if (k + 16 < K) {                      // warm next chunk (L2 -> WGP$)
            const float* Bn = Bp + bChunk;
#pragma unroll
            for (int j = 0; j < NT; ++j) __builtin_prefetch(Bn + j * 16, 0, 0);
            __builtin_prefetch(Ap + 16, 0, 0);
        }
#pragma unroll
        for (int ks = 0; ks < 4; ++ks) {
#pragma unroll
            for (int j = 0; j < NT; ++j) {
                acc[j] = __builtin_amdgcn_wmma_f32_16x16x4_f32(
                    false, a[ks], false, bfr[ks][j], (short)0, acc[j], false, false);
            }
        }
        Ap += 16;
        Bp += bChunk;
    }

    int rbase = m0 + ((lane >> 4) << 3);
#pragma unroll
    for (int j = 0; j < NT; ++j) {
#pragma unroll
        for (int r = 0; r < 8; ++r) {
            int col = n0 + j * 16 + mr;
            size_t off = (size_t)(rbase + r) * ldd + col;
            float v = acc[j][r];
            if (EPI == 1) v += bias[col];
            if (EPI == 2) { v += bias[col]; v = (v > 20.f) ? v : log1pf(expf(v)); }
            if (EPI == 3) v += Cin[off];
            D[off] = v;
        }
    }
}

// ---------------------------------------------------------------------------
// LayerNorm over width 768: one wave32 per row, 24 elems/lane, shuffle reduce.
// ---------------------------------------------------------------------------
__global__ void ln768(const float* __restrict__ in, const float* __restrict__ g,
                      const float* __restrict__ b, float* __restrict__ out, int rows)
{
    int gw = (int)((blockIdx.x * blockDim.x + threadIdx.x) >> 5);
    if (gw >= rows) return;
    int lane = threadIdx.x & 31;
    const float* p = in + (size_t)gw * DM_;
    float v[24];
    float s = 0.f;
#pragma unroll
    for (int i = 0; i < 24; ++i) { v[i] = p[lane + i * 32]; s += v[i]; }
#pragma unroll
    for (int o = 16; o > 0; o >>= 1) s += __shfl_xor(s, o, 32);
    float mean = s * (1.f / DM_);
    float ss = 0.f;
#pragma unroll
    for (int i = 0; i < 24; ++i) { float d = v[i] - mean; ss += d * d; }
#pragma unroll
    for (int o = 16; o > 0; o >>= 1) ss += __shfl_xor(ss, o, 32);
    float rstd = rsqrtf(ss * (1.f / DM_) + 1e-5f);
    float* q = out + (size_t)gw * DM_;
#pragma unroll
    for (int i = 0; i < 24; ++i) {
        int c = lane + i * 32;
        q[c] = (v[i] - mean) * rstd * g[c] + b[c];
    }
}

// ---------------------------------------------------------------------------
// Causal depthwise conv (DC=4) over L + SiLU. xc = first DI cols of xz.
// ---------------------------------------------------------------------------
__global__ void conv_silu(const float* __restrict__ xz, const float* __restrict__ cw,
                          const float* __restrict__ cb, float* __restrict__ u, int total)
{
    int idx = (int)(blockIdx.x * blockDim.x + threadIdx.x);
    if (idx >= total) return;
    int d  = idx % DI_;
    int bl = idx / DI_;
    int l  = bl % L_;
    int b  = bl / L_;
    float acc = cb[d];
    const float* w = cw + d * DC_;
#pragma unroll
    for (int j = 0; j < DC_; ++j) {
        int ls = l - (DC_ - 1) + j;
        if (ls >= 0) acc += xz[(size_t)(b * L_ + ls) * XZW + d] * w[j];
    }
    u[idx] = acc * sigmoidf_(acc);
}

// ---------------------------------------------------------------------------
// Selective scan, software-pipelined: loads for step l+1 are issued before
// the exp/fma chain of step l, hiding global latency behind the recurrence.
// 16 lanes = one channel's 16 states; 16 channels per 256-thread block.
// Fused epilogue: y = (scan + u*Dv) * silu(z).
// ---------------------------------------------------------------------------
__global__ void scan_fused(const float* __restrict__ delta, const float* __restrict__ u,
                           const float* __restrict__ dbc,   const float* __restrict__ xz,
                           const float* __restrict__ A_log, const float* __restrict__ Dv,
                           float* __restrict__ y)
{
    int g  = threadIdx.x >> 4;            // channel within block: 0..15
    int s  = threadIdx.x & 15;            // state index: 0..15
    int cb = blockIdx.x % (DI_ / 16);     // 0..95
    int b  = blockIdx.x / (DI_ / 16);
    int d  = cb * 16 + g;

    float a  = -expf(A_log[d * DS_ + s]);
    float dv = Dv[d];
    float st = 0.f;

    size_t bl0 = (size_t)b * L_;
    float dlt = delta[bl0 * DI_ + d];
    float uu  = u[bl0 * DI_ + d];
    float bb  = dbc[bl0 * DBW + DR_ + s];
    float cc  = dbc[bl0 * DBW + DR_ + DS_ + s];
    float zz  = xz[bl0 * XZW + DI_ + d];

    for (int l = 0; l < L_; ++l) {
        size_t bln = (size_t)b * L_ + l + 1;
        float ndlt = 0.f, nuu = 0.f, nbb = 0.f, ncc = 0.f, nzz = 0.f;
        if (l + 1 < L_) {                 // issue next-step loads early
            ndlt = delta[bln * DI_ + d];
            nuu  = u[bln * DI_ + d];
            nbb  = dbc[bln * DBW + DR_ + s];
            ncc  = dbc[bln * DBW + DR_ + DS_ + s];
            nzz  = xz[bln * XZW + DI_ + d];
        }

        st = st * expf(dlt * a) + (dlt * uu) * bb;
        float p = st * cc;
        p += __shfl_xor(p, 8, 16);
        p += __shfl_xor(p, 4, 16);
        p += __shfl_xor(p, 2, 16);
        p += __shfl_xor(p, 1, 16);
        if (s == 0) {
            size_t bl = (size_t)b * L_ + l;
            y[bl * DI_ + d] = (p + uu * dv) * (zz * sigmoidf_(zz));
        }

        dlt = ndlt; uu = nuu; bb = nbb; cc = ncc; zz = nzz;
    }
}

// ---------------------------------------------------------------------------
extern "C" void kernel_launch(void* const* d_in, const int* in_sizes, int n_in,
                              void* d_out, int out_size, void* d_ws, size_t ws_size,
                              hipStream_t stream)
{
    (void)in_sizes; (void)n_in; (void)out_size; (void)ws_size;
    const float* x       = (const float*)d_in[0];
    const float* Wi      = (const float*)d_in[1];
    const float* bi      = (const float*)d_in[2];
    const float* ln_g    = (const float*)d_in[3];
    const float* ln_b    = (const float*)d_in[4];
    const float* in_w    = (const float*)d_in[5];
    const float* conv_w  = (const float*)d_in[6];
    const float* conv_b  = (const float*)d_in[7];
    const float* xproj_w = (const float*)d_in[8];
    const float* dt_w    = (const float*)d_in[9];
    const float* dt_b    = (const float*)d_in[10];
    const float* A_log   = (const float*)d_in[11];
    const float* Dv      = (const float*)d_in[12];
    const float* out_w   = (const float*)d_in[13];
    const float* fn_g    = (const float*)d_in[14];
    const float* fn_b    = (const float*)d_in[15];

    float* ws    = (float*)d_ws;
    float* h     = ws;
    float* hn    = h     + (size_t)ROWS * DM_;
    float* xz    = hn    + (size_t)ROWS * DM_;
    float* u     = xz    + (size_t)ROWS * XZW;
    float* dbc   = u     + (size_t)ROWS * DI_;
    float* delta = dbc   + (size_t)ROWS * DBW;
    float* y     = delta + (size_t)ROWS * DI_;

    const int tilesM = ROWS / 16;   // 256

    // h = x @ Wi + bi   (M=4096, K=128, N=768)
    {
        int groups = DM_ / 64, tw = tilesM * groups;
        gemm_wmma_f32<4,1><<<(tw + 7) / 8, 256, 0, stream>>>(
            x, Wi, bi, nullptr, h, F_, F_, DM_, DM_, tilesM, tw);
    }

    for (int i = 0; i < NL_; ++i) {
        // hn = LN(h)
        ln768<<<ROWS / 8, 256, 0, stream>>>(h, ln_g + i * DM_, ln_b + i * DM_, hn, ROWS);

        // xz = hn @ in_w[i]   (K=768, N=3072)
        {
            int groups = XZW / 64, tw = tilesM * groups;
            gemm_wmma_f32<4,0><<<(tw + 7) / 8, 256, 0, stream>>>(
                hn, in_w + (size_t)i * DM_ * XZW, nullptr, nullptr, xz,
                DM_, DM_, XZW, XZW, tilesM, tw);
        }

        // u = silu(depthwise_conv(xc) + conv_b)
        conv_silu<<<(ROWS * DI_ + 255) / 256, 256, 0, stream>>>(
            xz, conv_w + (size_t)i * DI_ * DC_, conv_b + i * DI_, u, ROWS * DI_);

        // dbc = u @ xproj_w[i]   (K=1536, N=80)
        {
            int groups = DBW / 16, tw = tilesM * groups;
            gemm_wmma_f32<1,0><<<(tw + 7) / 8, 256, 0, stream>>>(
                u, xproj_w + (size_t)i * DI_ * DBW, nullptr, nullptr, dbc,
                DI_, DI_, DBW, DBW, tilesM, tw);
        }

        // delta = softplus(dt @ dt_w[i] + dt_b[i])  (A = first 48 cols of dbc)
        {
            int groups = DI_ / 64, tw = tilesM * groups;
            gemm_wmma_f32<4,2><<<(tw + 7) / 8, 256, 0, stream>>>(
                dbc, dt_w + (size_t)i * DR_ * DI_, dt_b + i * DI_, nullptr, delta,
                DR_, DBW, DI_, DI_, tilesM, tw);
        }

        // y = fused(selective scan, +u*Dv, *silu(z))
        scan_fused<<<B_ * (DI_ / 16), 256, 0, stream>>>(
            delta, u, dbc, xz, A_log + (size_t)i * DI_ * DS_, Dv + i * DI_, y);

        // h += y @ out_w[i]   (K=1536, N=768)
        {
            int groups = DM_ / 64, tw = tilesM * groups;
            gemm_wmma_f32<4,3><<<(tw + 7) / 8, 256, 0, stream>>>(
                y, out_w + (size_t)i * DI_ * DM_, nullptr, h, h,
                DI_, DI_, DM_, DM_, tilesM, tw);
        }
    }

    // out = LN(h) with fn_g / fn_b
    ln768<<<ROWS / 8, 256, 0, stream>>>(h, fn_g, fn_b, (float*)d_out, ROWS);
}